// GNN_65369402245678
// MI455X (gfx1250) — compile-verified
//
#include <hip/hip_runtime.h>
#include <hip/hip_bf16.h>
#include <math.h>

typedef __attribute__((ext_vector_type(2))) float v2f;
typedef __attribute__((ext_vector_type(8))) float v8f;

#define HC 32          // heads * channels
#define HEADS 2
#define CH 16
#define NEG_SLOPE 0.2f

// ---------------------------------------------------------------------------
// small init kernels
// ---------------------------------------------------------------------------
__global__ void zero_f(float* __restrict__ p, int n) {
    int i = blockIdx.x * blockDim.x + threadIdx.x;
    if (i < n) p[i] = 0.0f;
}

__global__ void init_attn_state(float* __restrict__ m, float* __restrict__ denom, int n2) {
    int i = blockIdx.x * blockDim.x + threadIdx.x;
    if (i < n2) { m[i] = -INFINITY; denom[i] = 0.0f; }
}

__global__ void init_out_bias(float* __restrict__ out, const float* __restrict__ bias, int total) {
    int i = blockIdx.x * blockDim.x + threadIdx.x;
    if (i < total) out[i] = bias[i & (HC - 1)];
}

// deg / self-loop edge_attr (fill='mean'): loopsum[d] += eattr[e]; deg[d] += 1
__global__ void deg_accum(const int* __restrict__ dst, const float* __restrict__ eattr,
                          float* __restrict__ deg, float* __restrict__ loopsum, int E) {
    int e = blockIdx.x * blockDim.x + threadIdx.x;
    if (e >= E) return;
    int d = dst[e];
    atomicAdd(&deg[d], 1.0f);
    atomicAdd(&loopsum[d], eattr[e]);
}

__global__ void finalize_loop(float* __restrict__ loopsum, const float* __restrict__ deg, int n) {
    int i = blockIdx.x * blockDim.x + threadIdx.x;
    if (i < n) loopsum[i] = loopsum[i] / fmaxf(deg[i], 1.0f);
}

// ---------------------------------------------------------------------------
// Dense node projection: Y = X @ W + b  (optional relu), W is K x 32
// One 16-row M-tile per wave32; two 16x16 N-tiles; K-loop step 4 via
// V_WMMA_F32_16X16X4_F32 (exact fp32 FMA on the matrix pipe).
// ---------------------------------------------------------------------------
__global__ void node_linear_wmma(const float* __restrict__ X, const float* __restrict__ W,
                                 const float* __restrict__ bias, float* __restrict__ Y,
                                 int n, int K, int relu) {
    __shared__ float sW[128 * HC];  // max K=128 -> 16KB
    int tid = threadIdx.x;
    int total = K * HC;
    for (int i = tid; i < total; i += blockDim.x) sW[i] = W[i];
    __syncthreads();

    int wave = tid >> 5;
    int lane = tid & 31;
    int mtiles = (n + 15) >> 4;
    int mtile = blockIdx.x * (blockDim.x >> 5) + wave;
    if (mtile >= mtiles) return;               // wave-uniform: EXEC stays all-1s

    int half = lane >> 4;                      // 0: K pair {k0,k0+1}, 1: {k0+2,k0+3}
    int lm = lane & 15;
    int m0 = mtile << 4;
    int row = m0 + lm;
    int rowc = row < n ? row : n - 1;          // clamp tail reads
    const float* xrow = X + (size_t)rowc * K;

    v8f acc0 = {};
    v8f acc1 = {};
#pragma unroll 4
    for (int k0 = 0; k0 < K; k0 += 4) {
        v2f a;
        a.x = xrow[k0 + 2 * half];
        a.y = xrow[k0 + 2 * half + 1];
        const float* w0 = &sW[(k0 + 2 * half) * HC];
        const float* w1 = &sW[(k0 + 2 * half + 1) * HC];
        v2f b0, b1;
        b0.x = w0[lm];       b0.y = w1[lm];
        b1.x = w0[16 + lm];  b1.y = w1[16 + lm];
        acc0 = __builtin_amdgcn_wmma_f32_16x16x4_f32(false, a, false, b0, (short)0, acc0, false, false);
        acc1 = __builtin_amdgcn_wmma_f32_16x16x4_f32(false, a, false, b1, (short)0, acc1, false, false);
    }

    // D layout: VGPR r holds rows (m0+r) on lanes 0-15 and (m0+r+8) on lanes 16-31
#pragma unroll
    for (int r = 0; r < 8; ++r) {
        int orow = m0 + r + 8 * half;
        if (orow >= n) continue;
        float v0 = acc0[r] + bias[lm];
        float v1 = acc1[r] + bias[16 + lm];
        if (relu) { v0 = fmaxf(v0, 0.0f); v1 = fmaxf(v1, 0.0f); }
        Y[(size_t)orow * HC + lm] = v0;
        Y[(size_t)orow * HC + 16 + lm] = v1;
    }
}

// ---------------------------------------------------------------------------
// Edge passes (memory bound; ~2-3 GB total traffic dominates runtime)
// ---------------------------------------------------------------------------
__device__ __forceinline__ void atomicMaxFloat(float* addr, float val) {
    if (val >= 0.0f) atomicMax((int*)addr, __float_as_int(val));
    else             atomicMin((unsigned int*)addr, __float_as_uint(val));
}

// pass 1: logits + segment max
__global__ void edge_logits_max(const int* __restrict__ src, const int* __restrict__ dst,
                                const float* __restrict__ eattr, const float* __restrict__ loopa,
                                const float* __restrict__ XL, const float* __restrict__ XR,
                                const float* __restrict__ We, const float* __restrict__ att,
                                float* __restrict__ logit, float* __restrict__ mbuf, int E, int n) {
    int e = blockIdx.x * blockDim.x + threadIdx.x;
    int E2 = E + n;
    if (e >= E2) return;
    int s, d; float ea;
    if (e < E) { s = src[e]; d = dst[e]; ea = eattr[e]; }
    else       { s = d = e - E;          ea = loopa[e - E]; }
    const float* xls = XL + (size_t)s * HC;
    const float* xrd = XR + (size_t)d * HC;
#pragma unroll
    for (int h = 0; h < HEADS; ++h) {
        float acc = 0.0f;
#pragma unroll
        for (int c = 0; c < CH; ++c) {
            int j = h * CH + c;
            float v = xls[j] + xrd[j] + ea * We[j];
            v = (v >= 0.0f) ? v : NEG_SLOPE * v;
            acc = fmaf(v, att[j], acc);
        }
        logit[(size_t)e * HEADS + h] = acc;
        atomicMaxFloat(&mbuf[(size_t)d * HEADS + h], acc);
    }
}

// pass 2: a = exp(logit - m[dst]); denom[dst] += a  (a overwrites logit in place)
__global__ void edge_exp_sum(const int* __restrict__ dst, float* __restrict__ logit,
                             const float* __restrict__ mbuf, float* __restrict__ denom, int E, int n) {
    int e = blockIdx.x * blockDim.x + threadIdx.x;
    int E2 = E + n;
    if (e >= E2) return;
    int d = (e < E) ? dst[e] : (e - E);
#pragma unroll
    for (int h = 0; h < HEADS; ++h) {
        float a = expf(logit[(size_t)e * HEADS + h] - mbuf[(size_t)d * HEADS + h]);
        logit[(size_t)e * HEADS + h] = a;
        atomicAdd(&denom[(size_t)d * HEADS + h], a);
    }
}

// pass 3: out[dst] += (a/denom[dst]) * xl[src]
__global__ void edge_scatter(const int* __restrict__ src, const int* __restrict__ dst,
                             const float* __restrict__ logit, const float* __restrict__ denom,
                             const float* __restrict__ XL, float* __restrict__ out, int E, int n) {
    int e = blockIdx.x * blockDim.x + threadIdx.x;
    int E2 = E + n;
    if (e >= E2) return;
    int s, d;
    if (e < E) { s = src[e]; d = dst[e]; }
    else       { s = d = e - E; }
    const float* xls = XL + (size_t)s * HC;
    float* od = out + (size_t)d * HC;
#pragma unroll
    for (int h = 0; h < HEADS; ++h) {
        float alpha = logit[(size_t)e * HEADS + h] / denom[(size_t)d * HEADS + h];
#pragma unroll
        for (int c = 0; c < CH; ++c)
            atomicAdd(&od[h * CH + c], alpha * xls[h * CH + c]);
    }
}

// decoder tail: out = hd @ Wd2 + bd2  (32 -> 2)
__global__ void decoder_out(const float* __restrict__ hd, const float* __restrict__ Wd2,
                            const float* __restrict__ bd2, float* __restrict__ out, int n) {
    int i = blockIdx.x * blockDim.x + threadIdx.x;
    if (i >= n) return;
    const float* hv = hd + (size_t)i * HC;
    float o0 = bd2[0], o1 = bd2[1];
#pragma unroll
    for (int k = 0; k < HC; ++k) {
        float h = hv[k];
        o0 = fmaf(h, Wd2[k * 2 + 0], o0);
        o1 = fmaf(h, Wd2[k * 2 + 1], o1);
    }
    out[(size_t)i * 2 + 0] = o0;
    out[(size_t)i * 2 + 1] = o1;
}

// ---------------------------------------------------------------------------
extern "C" void kernel_launch(void* const* d_in, const int* in_sizes, int n_in,
                              void* d_out, int out_size, void* d_ws, size_t ws_size,
                              hipStream_t stream) {
    const float* x     = (const float*)d_in[0];
    const int*   eidx  = (const int*)d_in[1];
    const float* eattr = (const float*)d_in[2];
    const float* Wl1 = (const float*)d_in[3];  const float* bl1 = (const float*)d_in[4];
    const float* Wr1 = (const float*)d_in[5];  const float* br1 = (const float*)d_in[6];
    const float* We1 = (const float*)d_in[7];  const float* att1 = (const float*)d_in[8];
    const float* b1  = (const float*)d_in[9];
    const float* Wl2 = (const float*)d_in[10]; const float* bl2 = (const float*)d_in[11];
    const float* Wr2 = (const float*)d_in[12]; const float* br2 = (const float*)d_in[13];
    const float* We2 = (const float*)d_in[14]; const float* att2 = (const float*)d_in[15];
    const float* b2  = (const float*)d_in[16];
    const float* Wd1 = (const float*)d_in[17]; const float* bd1 = (const float*)d_in[18];
    const float* Wd2 = (const float*)d_in[19]; const float* bd2 = (const float*)d_in[20];

    const int NODE_DIM = 128;
    int N = in_sizes[0] / NODE_DIM;
    int E = in_sizes[2];            // edge_attr has EDGE_DIM=1
    const int* src = eidx;
    const int* dst = eidx + E;
    int E2 = E + N;

    // workspace carve-out (floats)
    float* ws = (float*)d_ws;
    size_t off = 0;
    float* xl    = ws + off; off += (size_t)N * HC;
    float* xr    = ws + off; off += (size_t)N * HC;
    float* hbuf  = ws + off; off += (size_t)N * HC;   // layer1 out / layer2 in / layer2 out
    float* logit = ws + off; off += (size_t)E2 * HEADS;
    float* mbuf  = ws + off; off += (size_t)N * HEADS;
    float* denom = ws + off; off += (size_t)N * HEADS;
    float* deg   = ws + off; off += (size_t)N;
    float* loopa = ws + off; off += (size_t)N;        // loopsum -> loop_attr (in place)

    const int T = 256;
    int nb_N   = (N + T - 1) / T;
    int nb_2N  = (2 * N + T - 1) / T;
    int nb_E   = (E + T - 1) / T;
    int nb_E2  = (E2 + T - 1) / T;
    int nb_out = (N * HC + T - 1) / T;
    int mtiles = (N + 15) / 16;
    int nb_mm  = (mtiles + 7) / 8;   // 8 waves (M-tiles) per 256-thread block

    // --- self-loop edge_attr = mean of incoming edge_attr (shared by both layers) ---
    zero_f<<<nb_N, T, 0, stream>>>(deg, N);
    zero_f<<<nb_N, T, 0, stream>>>(loopa, N);
    deg_accum<<<nb_E, T, 0, stream>>>(dst, eattr, deg, loopa, E);
    finalize_loop<<<nb_N, T, 0, stream>>>(loopa, deg, N);

    auto run_layer = [&](const float* X, int K,
                         const float* Wl, const float* bl, const float* Wr, const float* br,
                         const float* We, const float* att, const float* bias, float* out) {
        node_linear_wmma<<<nb_mm, T, 0, stream>>>(X, Wl, bl, xl, N, K, 0);
        node_linear_wmma<<<nb_mm, T, 0, stream>>>(X, Wr, br, xr, N, K, 0);
        init_attn_state<<<nb_2N, T, 0, stream>>>(mbuf, denom, 2 * N);
        init_out_bias<<<nb_out, T, 0, stream>>>(out, bias, N * HC);
        edge_logits_max<<<nb_E2, T, 0, stream>>>(src, dst, eattr, loopa, xl, xr, We, att,
                                                 logit, mbuf, E, N);
        edge_exp_sum<<<nb_E2, T, 0, stream>>>(dst, logit, mbuf, denom, E, N);
        edge_scatter<<<nb_E2, T, 0, stream>>>(src, dst, logit, denom, xl, out, E, N);
    };

    run_layer(x,    128, Wl1, bl1, Wr1, br1, We1, att1, b1, hbuf);   // layer 1
    run_layer(hbuf,  32, Wl2, bl2, Wr2, br2, We2, att2, b2, hbuf);   // layer 2 (safe: stream-ordered)

    // decoder: hd = relu(h @ Wd1 + bd1) -> reuse xl buffer; then 32->2
    node_linear_wmma<<<nb_mm, T, 0, stream>>>(hbuf, Wd1, bd1, xl, N, 32, 1);
    decoder_out<<<nb_N, T, 0, stream>>>(xl, Wd2, bd2, (float*)d_out, N);
}